// GATEncoder_37726992728722
// MI455X (gfx1250) — compile-verified
//
#include <hip/hip_runtime.h>

#define D 64
#define LDSS 68   // padded LDS stride (floats): bank-conflict free, 8B aligned rows

typedef float v2f __attribute__((ext_vector_type(2)));
typedef float v8f __attribute__((ext_vector_type(8)));

// ---- order-preserving float <-> uint key for atomic max over signed floats ----
__device__ __forceinline__ unsigned f2key(float f) {
    unsigned u = __float_as_uint(f);
    return (u & 0x80000000u) ? ~u : (u | 0x80000000u);
}
__device__ __forceinline__ float key2f(unsigned k) {
    return (k & 0x80000000u) ? __uint_as_float(k & 0x7FFFFFFFu) : __uint_as_float(~k);
}

// H = X(N,64) @ W(64,64)  via v_wmma_f32_16x16x4_f32, with fused
// alpha_s = H @ a_s, alpha_d = H @ a_d computed in-register (shfl reductions).
// One wave per 16-row tile; 8 waves (tiles) per block. Requires N % 16 == 0.
__global__ __launch_bounds__(256) void gat_gemm_alpha(
    const float* __restrict__ X, const float* __restrict__ W,
    const float* __restrict__ a_s, const float* __restrict__ a_d,
    float* __restrict__ H, float* __restrict__ alpha_s, float* __restrict__ alpha_d,
    int nTiles)
{
    __shared__ float Ws[D * LDSS];          // transposed: Ws[col*LDSS + k]
    __shared__ float Xs[8 * 16 * LDSS];     // per-wave 16x64 A tile
    const int tid  = threadIdx.x;
    const int wave = tid >> 5;
    const int lane = tid & 31;
    const int tile = blockIdx.x * 8 + wave;

    // stage W transposed into LDS (whole block)
    for (int i = tid; i < D * D; i += 256) {
        int k = i >> 6, c = i & 63;
        Ws[c * LDSS + k] = W[i];
    }
    // stage this wave's 16x64 X tile (coalesced float4 reads)
    float* Xw = &Xs[wave * 16 * LDSS];
    if (tile < nTiles) {
        const float* Xg = X + (size_t)tile * 16 * D;
        for (int i = lane; i < 16 * 16; i += 32) {
            int r = i >> 4, c4 = (i & 15) * 4;
            float4 v = *(const float4*)(Xg + r * D + c4);
            float* p = &Xw[r * LDSS + c4];
            p[0] = v.x; p[1] = v.y; p[2] = v.z; p[3] = v.w;
        }
    }
    __syncthreads();
    if (tile >= nTiles) return;

    const int half = lane >> 4;   // 0: lanes 0-15, 1: lanes 16-31
    const int l16  = lane & 15;
    const int kb   = half * 2;    // A/B fragment K base within a k-step

    v8f zero = {};
    v8f acc[4] = { zero, zero, zero, zero };

    #pragma unroll
    for (int k0 = 0; k0 < D; k0 += 4) {
        // A fragment: lane holds X[row=l16][k0+kb], X[row=l16][k0+kb+1]
        v2f a = *(const v2f*)&Xw[l16 * LDSS + k0 + kb];
        #pragma unroll
        for (int nt = 0; nt < 4; ++nt) {
            int col = nt * 16 + l16;
            // B fragment: lane holds W[k0+kb][col], W[k0+kb+1][col]
            v2f b = *(const v2f*)&Ws[col * LDSS + k0 + kb];
            acc[nt] = __builtin_amdgcn_wmma_f32_16x16x4_f32(
                false, a, false, b, (short)0, acc[nt], false, false);
        }
    }

    // per-lane slices of a_src / a_dst (columns nt*16 + l16)
    float asv[4], adv[4];
    #pragma unroll
    for (int nt = 0; nt < 4; ++nt) {
        asv[nt] = a_s[nt * 16 + l16];
        adv[nt] = a_d[nt * 16 + l16];
    }

    const int rowBase = tile * 16 + half * 8;
    #pragma unroll
    for (int j = 0; j < 8; ++j) {
        int row = rowBase + j;
        float ps = 0.f, pd = 0.f;
        #pragma unroll
        for (int nt = 0; nt < 4; ++nt) {
            float v = acc[nt][j];
            H[(size_t)row * D + nt * 16 + l16] = v;
            ps += v * asv[nt];
            pd += v * adv[nt];
        }
        // reduce across the 16 lanes of this half-wave
        #pragma unroll
        for (int off = 8; off >= 1; off >>= 1) {
            ps += __shfl_xor(ps, off, 32);
            pd += __shfl_xor(pd, off, 32);
        }
        if (l16 == 0) { alpha_s[row] = ps; alpha_d[row] = pd; }
    }
}

__device__ __forceinline__ void edge_sd(const int* __restrict__ ei, int E, int e,
                                        int& s, int& d) {
    if (e < E) { s = ei[e]; d = ei[E + e]; }
    else       { s = e - E; d = s; }
}

// pass 1: e = leakyrelu(as[src]+ad[dst]); segment max via atomicMax on keys
__global__ void gat_edge_max(const int* __restrict__ ei, int E, int Etot,
                             const float* __restrict__ as, const float* __restrict__ ad,
                             float* __restrict__ elog, unsigned* __restrict__ mk)
{
    int e = blockIdx.x * 256 + threadIdx.x;
    if (e >= Etot) return;
    int s, d; edge_sd(ei, E, e, s, d);
    float v = as[s] + ad[d];
    v = v > 0.f ? v : 0.2f * v;
    elog[e] = v;
    atomicMax(mk + d, f2key(v));
}

// pass 2: w = exp(e - max[dst]); segment sum via atomicAdd
__global__ void gat_edge_expsum(const int* __restrict__ ei, int E, int Etot,
                                const unsigned* __restrict__ mk,
                                float* __restrict__ elog, float* __restrict__ denom)
{
    int e = blockIdx.x * 256 + threadIdx.x;
    if (e >= Etot) return;
    int s, d; edge_sd(ei, E, e, s, d);
    float w = expf(elog[e] - key2f(mk[d]));
    elog[e] = w;
    atomicAdd(denom + d, w);
}

// pass 3: alpha = w / denom[dst]
__global__ void gat_edge_norm(const int* __restrict__ ei, int E, int Etot,
                              const float* __restrict__ denom, float* __restrict__ elog)
{
    int e = blockIdx.x * 256 + threadIdx.x;
    if (e >= Etot) return;
    int s, d; edge_sd(ei, E, e, s, d);
    elog[e] = elog[e] / denom[d];
}

// pass 4: out[dst] += alpha * h[src]; 64 threads per edge, coalesced
__global__ void gat_aggregate(const int* __restrict__ ei, int E, int Etot,
                              const float* __restrict__ H, const float* __restrict__ elog,
                              float* __restrict__ out)
{
    unsigned gid = blockIdx.x * 256u + threadIdx.x;
    unsigned e = gid >> 6;
    int f = gid & 63;
    if (e >= (unsigned)Etot) return;
    int s, d; edge_sd(ei, E, (int)e, s, d);
    float v = elog[e] * H[(size_t)s * D + f];
    atomicAdd(out + (size_t)d * D + f, v);
}

__global__ void gat_bias_elu(const float* __restrict__ in, const float* __restrict__ b,
                             float* __restrict__ out, int total)
{
    int i = blockIdx.x * 256 + threadIdx.x;
    if (i >= total) return;
    float v = in[i] + b[i & 63];
    out[i] = v > 0.f ? v : expm1f(v);
}

__global__ void gat_bias(const float* __restrict__ b, float* __restrict__ out, int total)
{
    int i = blockIdx.x * 256 + threadIdx.x;
    if (i >= total) return;
    out[i] += b[i & 63];
}

extern "C" void kernel_launch(void* const* d_in, const int* in_sizes, int n_in,
                              void* d_out, int out_size, void* d_ws, size_t ws_size,
                              hipStream_t stream) {
    const int*   ei  = (const int*)d_in[0];
    const float* emb = (const float*)d_in[1];
    const float* W1  = (const float*)d_in[2];
    const float* a1s = (const float*)d_in[3];
    const float* a1d = (const float*)d_in[4];
    const float* b1  = (const float*)d_in[5];
    const float* W2  = (const float*)d_in[6];
    const float* a2s = (const float*)d_in[7];
    const float* a2d = (const float*)d_in[8];
    const float* b2  = (const float*)d_in[9];

    const int E    = in_sizes[0] / 2;
    const int N    = in_sizes[1] / D;
    const int Etot = E + N;
    const size_t feat = (size_t)N * D;

    char* p = (char*)d_ws;
    float*    H    = (float*)p;    p += feat * sizeof(float);
    float*    AGG  = (float*)p;    p += feat * sizeof(float);
    float*    X2   = (float*)p;    p += feat * sizeof(float);
    float*    as   = (float*)p;    p += (size_t)N * sizeof(float);
    float*    ad   = (float*)p;    p += (size_t)N * sizeof(float);
    unsigned* mk   = (unsigned*)p; p += (size_t)N * sizeof(unsigned);
    float*    dn   = (float*)p;    p += (size_t)N * sizeof(float);
    float*    elog = (float*)p;    p += (size_t)Etot * sizeof(float);

    const int nTiles = N / 16;
    const unsigned gGemm = (unsigned)((nTiles + 7) / 8);
    const unsigned gE    = (unsigned)((Etot + 255) / 256);
    const unsigned gAgg  = (unsigned)(((size_t)Etot * 64 + 255) / 256);
    const unsigned gF    = (unsigned)((feat + 255) / 256);
    const int totalF = (int)feat;

    auto layer = [&](const float* x, const float* W, const float* a_s, const float* a_d,
                     const float* bias, float* accum, float* act_out, bool do_elu) {
        hipMemsetAsync(mk, 0, (size_t)N * sizeof(unsigned), stream);
        hipMemsetAsync(dn, 0, (size_t)N * sizeof(float), stream);
        hipMemsetAsync(accum, 0, feat * sizeof(float), stream);
        gat_gemm_alpha<<<gGemm, 256, 0, stream>>>(x, W, a_s, a_d, H, as, ad, nTiles);
        gat_edge_max   <<<gE,   256, 0, stream>>>(ei, E, Etot, as, ad, elog, mk);
        gat_edge_expsum<<<gE,   256, 0, stream>>>(ei, E, Etot, mk, elog, dn);
        gat_edge_norm  <<<gE,   256, 0, stream>>>(ei, E, Etot, dn, elog);
        gat_aggregate  <<<gAgg, 256, 0, stream>>>(ei, E, Etot, H, elog, accum);
        if (do_elu) gat_bias_elu<<<gF, 256, 0, stream>>>(accum, bias, act_out, totalF);
        else        gat_bias    <<<gF, 256, 0, stream>>>(bias, accum, totalF);
    };

    // layer 1: embedding -> X2 (ELU)
    layer(emb, W1, a1s, a1d, b1, AGG, X2, true);
    // layer 2: X2 -> d_out (bias only)
    layer(X2, W2, a2s, a2d, b2, (float*)d_out, nullptr, false);
}